// RoiAlign_25941602467885
// MI455X (gfx1250) — compile-verified
//
#include <hip/hip_runtime.h>

typedef float v2f __attribute__((ext_vector_type(2)));
typedef float v4f __attribute__((ext_vector_type(4)));
typedef float v8f __attribute__((ext_vector_type(8)));

#define POOL   7
#define CCHUNK 128
#define NCOL   (POOL * CCHUNK)   // 896 columns = (j, c-chunk)
#define NV4    (NCOL / 4)        // 224 float4 columns
#define KROWS  16                // 14 interpolated rows + 2 zero pad

__global__ __launch_bounds__(256) void roialign_wmma_kernel(
    const float* __restrict__ feat0,   // [2,256,256,256]
    const float* __restrict__ feat1,   // [2,128,128,256]
    const float* __restrict__ rois,    // [2,512,5] (y1,x1,y2,x2,0)
    float* __restrict__ out)           // [2,512,7,7,256]
{
    __shared__ __align__(16) float U[KROWS * NCOL];  // 57344 B

    const int bid  = blockIdx.x;       // box id 0..1023
    const int tid  = threadIdx.x;
    const int lane = tid & 31;
    const int wave = tid >> 5;

    const int   b  = bid >> 9;         // batch (R=512)
    const float y1 = rois[bid * 5 + 0];
    const float x1 = rois[bid * 5 + 1];
    const float y2 = rois[bid * 5 + 2];
    const float x2 = rois[bid * 5 + 3];

    const bool level = ((y2 - y1) > 48.0f) || ((x2 - x1) > 48.0f);
    const float* __restrict__ feat = level ? feat1 : feat0;
    const int   H   = level ? 128 : 256;
    const int   W   = H;
    const int   C   = 256;
    const float hm1 = (float)(H - 1);

    const float inv = 1.0f / 1024.0f;            // normalize by IMAGE_MAX_DIM
    const float yb = y1 * inv * hm1, yS = (y2 - y1) * inv * hm1;
    const float xb = x1 * inv * hm1, xS = (x2 - x1) * inv * hm1;

    const v4f* __restrict__ featB4 =
        (const v4f*)(feat + (size_t)b * H * W * C);          // float4 view

    // ---- per-lane y-weights for the WMMA A matrix (Wy is 7x14, 2 nnz/row) ----
    const int m = lane & 15;                     // M row this lane supplies
    float a0 = 0.0f, a1 = 0.0f;                  // coeffs at K = m and K = m+7
    if (m < POOL) {
        float ysm = yb + (float)m * (1.0f / 6.0f) * yS;
        float y0f = floorf(ysm);
        float wy  = ysm - y0f;
        bool  vy  = (ysm >= 0.0f) && (ysm <= hm1);
        a0 = vy ? (1.0f - wy) : 0.0f;
        a1 = vy ? wy : 0.0f;
    }

    v4f* __restrict__ U4 = (v4f*)U;

    for (int chunk = 0; chunk < 2; ++chunk) {
        const int cbase4 = chunk * (CCHUNK / 4);             // float4 channel base

        // ---- phase 1: x-interpolated rows into LDS, 16B-vectorized gathers ----
        for (int e = tid; e < 14 * NV4; e += 256) {
            int n   = e / NV4;
            int rem = e - n * NV4;                 // (j, c4): j*32 + c4
            int j   = rem >> 5;
            int c4  = cbase4 + (rem & 31);
            int i    = (n < POOL) ? n : (n - POOL);
            int half = (n < POOL) ? 0 : 1;

            float ysi = yb + (float)i * (1.0f / 6.0f) * yS;
            int   r   = (int)floorf(ysi) + half;
            r = min(max(r, 0), H - 1);

            float xsj = xb + (float)j * (1.0f / 6.0f) * xS;
            float x0f = floorf(xsj);
            float wx  = xsj - x0f;
            int   x0  = min(max((int)x0f, 0), W - 1);
            int   xp  = min(max((int)x0f + 1, 0), W - 1);
            bool  vx  = (xsj >= 0.0f) && (xsj <= (float)(W - 1));

            const v4f* rowp = featB4 + (size_t)r * W * (C / 4);
            v4f v = {};
            if (vx) {
                v4f p0 = rowp[x0 * (C / 4) + c4];
                v4f p1 = rowp[xp * (C / 4) + c4];
                v = p0 + wx * (p1 - p0);
            }
            U4[e] = v;                             // ds_store_b128
        }
        // zero K pad rows 14,15 so the last K=4 step contributes nothing
        for (int e = tid; e < 2 * NV4; e += 256)
            U4[14 * NV4 + e] = (v4f){};
        __syncthreads();

        // ---- phase 2: Out[i,(j,c)] = sum_n Wy[i,n]*U[n,(j,c)] via V_WMMA_F32_16X16X4_F32 ----
        const int koff = (lane >> 4) << 1;       // A/B layout: lanes 0-15 -> K+0/K+1, 16-31 -> K+2/K+3
        for (int t = wave; t < NCOL / 16; t += 8) {   // 56 tiles / 8 waves
            const int col = t * 16 + (lane & 15);
            v8f acc = {};
#pragma unroll
            for (int ks = 0; ks < 4; ++ks) {
                const int kb = ks * 4 + koff;
                v2f A;
                A.x = (kb     == m) ? a0 : ((kb     == m + POOL) ? a1 : 0.0f);
                A.y = (kb + 1 == m) ? a0 : ((kb + 1 == m + POOL) ? a1 : 0.0f);
                v2f Bv;
                Bv.x = U[(kb    ) * NCOL + col];
                Bv.y = U[(kb + 1) * NCOL + col];
                acc = __builtin_amdgcn_wmma_f32_16x16x4_f32(
                        false, A, false, Bv, (short)0, acc, false, false);
            }
            // D layout: VGPR v holds M = v (lanes 0-15) / v+8 (lanes 16-31); only M<7 valid
            const int j = col / CCHUNK;
            const int c = chunk * CCHUNK + (col - j * CCHUNK);
            if (lane < 16) {
                float* op = out + ((size_t)bid * POOL * POOL + j) * 256 + c;
#pragma unroll
                for (int v = 0; v < POOL; ++v)
                    op[(size_t)v * POOL * 256] = acc[v];
            }
        }
        __syncthreads();
    }
}

extern "C" void kernel_launch(void* const* d_in, const int* in_sizes, int n_in,
                              void* d_out, int out_size, void* d_ws, size_t ws_size,
                              hipStream_t stream) {
    const float* feat0 = (const float*)d_in[0];
    const float* feat1 = (const float*)d_in[1];
    const float* rois  = (const float*)d_in[2];
    float* out = (float*)d_out;
    roialign_wmma_kernel<<<dim3(1024), dim3(256), 0, stream>>>(feat0, feat1, rois, out);
}